// KalmanFilter_90709709291981
// MI455X (gfx1250) — compile-verified
//
#include <hip/hip_runtime.h>
#include <hip/hip_bf16.h>

// ---------------------------------------------------------------------------
// Batched Kalman filter, MI455X (gfx1250, wave32).
// 4 groups per wave; the 4 groups' 8x8 covariances are packed into one 16x16
// V_WMMA_F32_16X16X4_F32 C/D tile:
//     [[P_g0, P_g1],
//      [P_g2, P_g3]]
// C/D layout fact used throughout: VGPR r, lane l  <->  row (r + 8*(l>=16)),
// col (l % 16).  Hence lane 8g+s holds COLUMN s of group g's block across the
// 8 tile VGPRs -> H*P, innovation covariance, 2x2 solve are per-lane math with
// ds_swizzle 8-lane reductions.  The heavy 8x8 products run on the matrix
// pipe as chained fp32 WMMAs.
// ---------------------------------------------------------------------------

typedef float v2f __attribute__((ext_vector_type(2)));
typedef float v8f __attribute__((ext_vector_type(8)));

template <int PAT>
__device__ __forceinline__ float swz(float v) {
  return __int_as_float(__builtin_amdgcn_ds_swizzle(__float_as_int(v), PAT));
}

// sum over each aligned 8-lane group (result replicated to all 8 lanes)
__device__ __forceinline__ float red8(float v) {
  v += swz<0x041F>(v);  // xor 1
  v += swz<0x081F>(v);  // xor 2
  v += swz<0x101F>(v);  // xor 4
  return v;
}

__device__ __forceinline__ v8f wmma4(v2f a, v2f b, v8f c) {
  // D = A(16x4,f32) * B(4x16,f32) + C(16x16,f32)
  return __builtin_amdgcn_wmma_f32_16x16x4_f32(false, a, false, b, (short)0, c,
                                               false, false);
}

__global__ __launch_bounds__(256) void kf_tile_kernel(
    const float* __restrict__ x, const float* __restrict__ m0,
    const float* __restrict__ P0, const float* __restrict__ Fm,
    const float* __restrict__ Qm, const float* __restrict__ Hm,
    const float* __restrict__ Rm, float* __restrict__ out, int G, int T) {
  __shared__ float lds[8 * 256];  // one 16x16 f32 scratch tile per wave

  const int lane  = threadIdx.x & 31;
  const int wgl   = threadIdx.x >> 5;
  const int wave  = blockIdx.x * (blockDim.x >> 5) + wgl;
  if (wave * 4 >= G) return;  // uniform per wave: EXEC stays all-ones for WMMA
  const int g     = wave * 4 + (lane >> 3);  // this lane's group
  const int s     = lane & 7;                // state/column index
  const int l16   = lane & 15;
  const bool lo16 = lane < 16;
  const bool q0 = lane < 8;
  const bool q1 = (lane >= 8) && lo16;
  const bool q2 = (lane >= 16) && (lane < 24);
  const bool q3 = lane >= 24;

  float* tile = &lds[wgl * 256];

  // ---- uniform constants (H, R) -> SGPR-resident scalars
  float Hs[16];
#pragma unroll
  for (int i = 0; i < 16; ++i) Hs[i] = Hm[i];
  const float r00 = Rm[0], r01 = Rm[1], r11 = Rm[3];

  // ---- per-lane constants
  const float hv0 = Hm[s], hv1 = Hm[8 + s];  // H[0][s], H[1][s]
  float Frow[8], Qt[8];
#pragma unroll
  for (int k = 0; k < 8; ++k) {
    Frow[k] = Fm[s * 8 + k];  // F[s][k]   (m' = F m)
    Qt[k]   = Qm[k * 8 + s];  // Q tile in C/D layout (all 4 blocks get +Q)
  }
  // blockdiag(F,F) in WMMA A layout; 4 K-chunks of 4 (A chunk: VGPR0 holds
  // K = 4c / 4c+2 for lanes <16 / >=16, VGPR1 holds 4c+1 / 4c+3)
  v2f FA[4];
#pragma unroll
  for (int c = 0; c < 4; ++c) {
    const int kx   = 4 * c + (lo16 ? 0 : 2);
    const float fx = Fm[(l16 & 7) * 8 + (kx & 7)];
    const float fy = Fm[(l16 & 7) * 8 + ((kx + 1) & 7)];
    const bool on  = ((l16 < 8) == (kx < 8));  // block-diagonal mask
    FA[c].x = on ? fx : 0.f;
    FA[c].y = on ? fy : 0.f;
  }

  // ---- state init: m in lane=8g+s layout, P in the packed C/D tile
  float mv = m0[(size_t)g * 8 + s];
  v8f Pt;
#pragma unroll
  for (int r = 0; r < 8; ++r) Pt[r] = P0[(size_t)g * 64 + r * 8 + s];

  const float2* xg = reinterpret_cast<const float2*>(x) + (size_t)g * T;
  float2* outm = reinterpret_cast<float2*>(out) + (size_t)g * T;
  float4* outc =
      reinterpret_cast<float4*>(out + (size_t)G * T * 2) + (size_t)g * T;

  // LDS addressing for the per-block transpose of X (stored row-major 16x16)
  const int st  = l16 + (lo16 ? 0 : 128);                      // store base
  const int ldb = (l16 & 7) * 16 + (l16 & 8) + (lo16 ? 0 : 2); // load base
  const int bb  = (lane & 24) << 2;  // bpermute byte base of this 8-lane group

  for (int t = 0; t < T; ++t) {
    // ---- HP = H * P  (lane holds column s of its group's P)
    float hp0 = 0.f, hp1 = 0.f;
#pragma unroll
    for (int k = 0; k < 8; ++k) {
      hp0 = fmaf(Hs[k],     Pt[k], hp0);
      hp1 = fmaf(Hs[8 + k], Pt[k], hp1);
    }
    // ---- innovation covariance S = HP H^T + R and predicted mean H m
    const float S00 = red8(hp0 * hv0) + r00;
    const float S01 = red8(hp0 * hv1) + r01;
    const float S11 = red8(hp1 * hv1) + r11;
    const float me0 = red8(hv0 * mv);
    const float me1 = red8(hv1 * mv);

    // ---- outputs for step t (pre-update prediction), one lane per group
    if (s == 0) {
      float2 mo; mo.x = me0; mo.y = me1;
      outm[t] = mo;                                   // global_store_b64
      float4 co; co.x = S00; co.y = S01; co.z = S01; co.w = S11;
      outc[t] = co;                                   // global_store_b128
    }

    // ---- gain Kt = S^{-1} HP  (2x2 closed-form inverse, per-lane)
    const float inv = 1.f / (S00 * S11 - S01 * S01);
    const float i00 = S11 * inv, i01 = -S01 * inv, i11 = S00 * inv;
    const float kt0 = i00 * hp0 + i01 * hp1;  // Kt[0][s]
    const float kt1 = i01 * hp0 + i11 * hp1;  // Kt[1][s]

    // ---- innovation + mean update
    const float2 xv = xg[t];
    __builtin_prefetch(&xg[(t + 8 < T) ? (t + 8) : (T - 1)], 0, 1);
    const float y0 = xv.x - me0, y1 = xv.y - me1;
    const float mu = fmaf(kt1, y1, fmaf(kt0, y0, mv));

    // ---- rank-2 downdate P_u = P - K*HP as two K=4 WMMAs, 4 groups packed
    const float nk0 = -kt0, nk1 = -kt1;
    const float tk0 = swz<0x601F>(nk0);  // lane ^ 24
    const float tk1 = swz<0x601F>(nk1);
    const float sh0 = swz<0x401F>(hp0);  // swap lane halves
    const float sh1 = swz<0x401F>(hp1);
    v2f A0, A1, B0, B1;
    A0.x = q0 ? nk0 : (q2 ? tk0 : 0.f);  // slots g0/g1, meas 0
    A0.y = q0 ? nk1 : (q2 ? tk1 : 0.f);  //              meas 1
    A1.x = q1 ? tk0 : (q3 ? nk0 : 0.f);  // slots g2/g3
    A1.y = q1 ? tk1 : (q3 ? nk1 : 0.f);
    B0.x = q0 ? hp0 : (q3 ? sh0 : 0.f);
    B0.y = q0 ? hp1 : (q3 ? sh1 : 0.f);
    B1.x = q0 ? sh0 : (q3 ? hp0 : 0.f);
    B1.y = q0 ? sh1 : (q3 ? hp1 : 0.f);
    Pt = wmma4(A0, B0, Pt);
    Pt = wmma4(A1, B1, Pt);  // Pt = P_u

    // ---- D-layout -> B-layout for P_u (8 swapx16 + selects)
    const float w0 = swz<0x401F>(Pt[0]); const float w1 = swz<0x401F>(Pt[1]);
    const float w2 = swz<0x401F>(Pt[2]); const float w3 = swz<0x401F>(Pt[3]);
    const float w4 = swz<0x401F>(Pt[4]); const float w5 = swz<0x401F>(Pt[5]);
    const float w6 = swz<0x401F>(Pt[6]); const float w7 = swz<0x401F>(Pt[7]);
    v2f Bp0, Bp1, Bp2, Bp3;
    Bp0.x = lo16 ? Pt[0] : w2;  Bp0.y = lo16 ? Pt[1] : w3;   // rows 0..3
    Bp1.x = lo16 ? Pt[4] : w6;  Bp1.y = lo16 ? Pt[5] : w7;   // rows 4..7
    Bp2.x = lo16 ? w0 : Pt[2];  Bp2.y = lo16 ? w1 : Pt[3];   // rows 8..11
    Bp3.x = lo16 ? w4 : Pt[6];  Bp3.y = lo16 ? w5 : Pt[7];   // rows 12..15

    // ---- X = blockdiag(F) @ P_u   (K=16 as 4 chained WMMAs)
    v8f X;
#pragma unroll
    for (int r = 0; r < 8; ++r) X[r] = 0.f;
    X = wmma4(FA[0], Bp0, X);
    X = wmma4(FA[1], Bp1, X);
    X = wmma4(FA[2], Bp2, X);
    X = wmma4(FA[3], Bp3, X);

    // ---- per-8x8-block transpose of X through LDS (keeps groups in place)
#pragma unroll
    for (int r = 0; r < 8; ++r) tile[st + r * 16] = X[r];
    asm volatile("s_wait_dscnt 0x0" ::: "memory");
    v2f Bx0, Bx1, Bx2, Bx3;
    Bx0.x = tile[ldb + 0];       Bx0.y = tile[ldb + 1];
    Bx1.x = tile[ldb + 4];       Bx1.y = tile[ldb + 5];
    Bx2.x = tile[ldb + 128 + 0]; Bx2.y = tile[ldb + 128 + 1];
    Bx3.x = tile[ldb + 128 + 4]; Bx3.y = tile[ldb + 128 + 5];

    // ---- P' = blockdiag(F) @ X^T_blk + Q  (C seeded with the Q tile)
    v8f Pn;
#pragma unroll
    for (int r = 0; r < 8; ++r) Pn[r] = Qt[r];
    Pn = wmma4(FA[0], Bx0, Pn);
    Pn = wmma4(FA[1], Bx1, Pn);
    Pn = wmma4(FA[2], Bx2, Pn);
    Pn = wmma4(FA[3], Bx3, Pn);
    Pt = Pn;

    // ---- m' = F @ m_u via 8-lane broadcasts (ds_bpermute)
    float mp = 0.f;
#pragma unroll
    for (int k = 0; k < 8; ++k) {
      const float bk = __int_as_float(
          __builtin_amdgcn_ds_bpermute(bb + (k << 2), __float_as_int(mu)));
      mp = fmaf(Frow[k], bk, mp);
    }
    mv = mp;
  }
}

extern "C" void kernel_launch(void* const* d_in, const int* in_sizes, int n_in,
                              void* d_out, int out_size, void* d_ws,
                              size_t ws_size, hipStream_t stream) {
  const float* x  = (const float*)d_in[0];
  const float* m0 = (const float*)d_in[1];
  const float* P0 = (const float*)d_in[2];
  const float* F  = (const float*)d_in[3];
  const float* Q  = (const float*)d_in[4];
  const float* H  = (const float*)d_in[5];
  const float* R  = (const float*)d_in[6];
  float* out = (float*)d_out;

  const int G = in_sizes[1] / 8;        // m0 is [G, 8]
  const int T = in_sizes[0] / (G * 2);  // x  is [G, T, 2]

  const int groups_per_block = 32;  // 8 waves * 4 groups
  dim3 grid((G + groups_per_block - 1) / groups_per_block);
  kf_tile_kernel<<<grid, 256, 0, stream>>>(x, m0, P0, F, Q, H, R, out, G, T);
}